// SparseNeighborhoodAttentionBlock_69655779607138
// MI455X (gfx1250) — compile-verified
//
#include <hip/hip_runtime.h>
#include <hip/hip_bf16.h>
#include <cstdint>

// ---------------------------------------------------------------------------
// SparseNeighborhoodAttentionBlock for MI455X (gfx1250, wave32, WMMA)
// ---------------------------------------------------------------------------
#define N_Q    1024
#define DIM    384
#define TWOD   768
#define NHD    8
#define HDIM   48
#define HHALF  24
#define NFREQ  8
#define NLVL   4
#define NSW    7
#define NKEY   (NLVL * NSW * NSW)   // 196
#define TOTAL  5440                  // 64*64 + 32*32 + 16*16 + 8*8
#define NBATCH 4
#define MROWS_KV (NBATCH * TOTAL)    // 21760

typedef float v2f __attribute__((ext_vector_type(2)));
typedef float v8f __attribute__((ext_vector_type(8)));

// gfx1250 async global->LDS path (ASYNCcnt-tracked), with portable fallback
#if defined(__AMDGCN__) && __has_builtin(__builtin_amdgcn_global_load_async_to_lds_b128)
#define HAVE_ASYNC_LDS 1
#else
#define HAVE_ASYNC_LDS 0
#endif

#if HAVE_ASYNC_LDS
typedef int ai4 __attribute__((vector_size(16)));
typedef __attribute__((address_space(1))) ai4* async_gp_t;   // global src
typedef __attribute__((address_space(3))) ai4* async_lp_t;   // LDS dst
#endif

template <int CNT>
__device__ __forceinline__ void wait_async_lds()
{
#if HAVE_ASYNC_LDS
#if __has_builtin(__builtin_amdgcn_s_wait_asynccnt)
    __builtin_amdgcn_s_wait_asynccnt(CNT);
#else
    asm volatile("s_wait_asynccnt %0" :: "i"(CNT) : "memory");
#endif
#endif
}

// ---------------------------------------------------------------------------
// Kernel 1: LayerNorm.  One wave32 per row (D=384 -> 12 elems/lane).
// ---------------------------------------------------------------------------
__global__ __launch_bounds__(256) void ln_kernel(
    const float* __restrict__ x, const float* __restrict__ w,
    const float* __restrict__ b, float* __restrict__ y)
{
    const int row  = blockIdx.x * 8 + (threadIdx.x >> 5);
    const int lane = threadIdx.x & 31;
    const float* xr = x + (size_t)row * DIM;

    float vals[12];
    float s = 0.f, ss = 0.f;
#pragma unroll
    for (int i = 0; i < 12; ++i) {
        float v = xr[lane + 32 * i];
        vals[i] = v; s += v; ss += v * v;
    }
#pragma unroll
    for (int off = 16; off; off >>= 1) {
        s  += __shfl_xor(s,  off, 32);
        ss += __shfl_xor(ss, off, 32);
    }
    const float mean = s * (1.f / DIM);
    const float var  = ss * (1.f / DIM) - mean * mean;
    const float inv  = rsqrtf(var + 1e-5f);

    float* yr = y + (size_t)row * DIM;
#pragma unroll
    for (int i = 0; i < 12; ++i) {
        int c = lane + 32 * i;
        yr[c] = (vals[i] - mean) * inv * w[c] + b[c];
    }
}

// ---------------------------------------------------------------------------
// Kernel 2: WMMA fp32 GEMM, Y[M,N] = X[M,K=384] @ W[N,384]^T (+ residual).
//
// Block: 256 threads = 8 waves.  Block tile: 128(M) x 64(N) x 384(K).
// The 64-row B strip is shared by all 8 waves, so it is staged in LDS once
// per block: 4 K-chunks of 96, double-buffered (2 x 64 x 100 floats, 50 KB),
// filled with GLOBAL_LOAD_ASYNC_TO_LDS_B128 (ASYNCcnt) overlapped with the
// WMMA compute on the previous chunk.  Row stride 100 floats keeps 16B
// alignment and a conflict-free bank pattern for the ds_load_b64 fragments.
//
// fp32 WMMA fragment layout (V_WMMA_F32_16X16X4_F32):
//   A(16x4): lane = m + 16*(k>>1), vgpr = k&1  -> one b64 load / lane
//   B(4x16): lane = n + 16*(k>>1), vgpr = k&1  -> one ds_load_b64 / lane
//   C/D:     vgpr r: M = r + 8*(lane>>4), N = lane&15
// M must be a multiple of 128 (true for all launches: 1024, 21760).
// ---------------------------------------------------------------------------
#define KCHUNK  96
#define NCHUNK  (DIM / KCHUNK)      // 4
#define BROWS   64
#define BSTRIDE 100                 // floats; 400 B row stride

__device__ __forceinline__ void stage_b_chunk(
    const float* __restrict__ W, int n0, int chunk, float* dst, int t)
{
    // 64 rows x 96 floats = 1536 x 16B granules; 6 per thread (256 threads)
#pragma unroll
    for (int i = 0; i < 6; ++i) {
        const int idx = t + i * 256;
        const int row = idx / 24;
        const int c4  = (idx % 24) * 4;
        const float* src = W + (size_t)(n0 + row) * DIM + chunk * KCHUNK + c4;
        float* d = dst + row * BSTRIDE + c4;
#if HAVE_ASYNC_LDS
        async_gp_t gs = (async_gp_t)(uintptr_t)src;            // AS1 (value-preserving)
        async_lp_t ls = (async_lp_t)(unsigned)(uintptr_t)d;    // low 32 bits = LDS offset
        __builtin_amdgcn_global_load_async_to_lds_b128(gs, ls, 0, 0);
#else
        float4 v = *(const float4*)src;
        *(float4*)d = v;
#endif
    }
}

__global__ __launch_bounds__(256) void wmma_gemm_nt_kernel(
    const float* __restrict__ X, const float* __restrict__ W,
    const float* __restrict__ R,   // residual (may be null)
    float* __restrict__ Y, int M, int Nout)
{
    __shared__ float sB[2][BROWS * BSTRIDE];

    const int t    = threadIdx.x;
    const int wave = t >> 5;
    const int lane = t & 31;
    const int half = lane >> 4;     // selects K pair {0,1} vs {2,3}
    const int lm   = lane & 15;

    const int m0 = (blockIdx.x * 8 + wave) * 16;
    const int n0 = blockIdx.y * 64;

    const float* aptr = X + (size_t)(m0 + lm) * DIM + 2 * half;

    v8f acc0 = {0.f,0.f,0.f,0.f,0.f,0.f,0.f,0.f};
    v8f acc1 = acc0, acc2 = acc0, acc3 = acc0;

    stage_b_chunk(W, n0, 0, &sB[0][0], t);

#pragma unroll
    for (int c = 0; c < NCHUNK; ++c) {
        const int cur = c & 1;
        if (c + 1 < NCHUNK) {
            stage_b_chunk(W, n0, c + 1, &sB[cur ^ 1][0], t);
            wait_async_lds<6>();     // chunk c complete; chunk c+1 in flight
        } else {
            wait_async_lds<0>();
        }
        __syncthreads();             // chunk c visible to all waves

        const float* a = aptr + c * KCHUNK;
        const float* b = &sB[cur][0] + lm * BSTRIDE + 2 * half;
#pragma unroll
        for (int k = 0; k < KCHUNK; k += 4) {
            v2f av = *(const v2f*)(a + k);
            v2f b0 = *(const v2f*)(b + 0  * 16 * BSTRIDE + k);
            v2f b1 = *(const v2f*)(b + 1  * 16 * BSTRIDE + k);
            v2f b2 = *(const v2f*)(b + 2  * 16 * BSTRIDE + k);
            v2f b3 = *(const v2f*)(b + 3  * 16 * BSTRIDE + k);
            acc0 = __builtin_amdgcn_wmma_f32_16x16x4_f32(false, av, false, b0, (short)0, acc0, false, false);
            acc1 = __builtin_amdgcn_wmma_f32_16x16x4_f32(false, av, false, b1, (short)0, acc1, false, false);
            acc2 = __builtin_amdgcn_wmma_f32_16x16x4_f32(false, av, false, b2, (short)0, acc2, false, false);
            acc3 = __builtin_amdgcn_wmma_f32_16x16x4_f32(false, av, false, b3, (short)0, acc3, false, false);
        }
        __syncthreads();             // done reading buf before it is refilled
    }

    const int mb = m0 + half * 8;
#pragma unroll
    for (int r = 0; r < 8; ++r) {
        const size_t rowb = (size_t)(mb + r) * Nout + n0 + lm;
        float v0 = acc0[r], v1 = acc1[r], v2 = acc2[r], v3 = acc3[r];
        if (R) {
            v0 += R[rowb +  0]; v1 += R[rowb + 16];
            v2 += R[rowb + 32]; v3 += R[rowb + 48];
        }
        Y[rowb +  0] = v0; Y[rowb + 16] = v1;
        Y[rowb + 32] = v2; Y[rowb + 48] = v3;
    }
}

// ---------------------------------------------------------------------------
// Kernel 3: sparse neighborhood attention.  One 384-thread block per query.
// ---------------------------------------------------------------------------
__global__ __launch_bounds__(384) void attn_kernel(
    const float* __restrict__ qproj,   // N_Q x DIM
    const float* __restrict__ kvmaps,  // MROWS_KV x TWOD
    const int*   __restrict__ pos,     // N_Q x 4 (b,i,j,l)
    const int*   __restrict__ shapes,  // NLVL x 2
    float* __restrict__ attnout)       // N_Q x DIM
{
    const int n = blockIdx.x;
    const int t = threadIdx.x;

    __shared__ float s_qrot[DIM];
    __shared__ float s_cos[NKEY * HHALF];
    __shared__ float s_sin[NKEY * HHALF];
    __shared__ int   s_flat[NKEY];
    __shared__ float s_mask[NKEY];
    __shared__ float s_kif[NKEY];
    __shared__ float s_kjf[NKEY];
    __shared__ float s_sc[NHD * NKEY];

    int Hs[NLVL], Wsh[NLVL], offs[NLVL];
    {
        int a = 0;
#pragma unroll
        for (int l = 0; l < NLVL; ++l) {
            Hs[l] = shapes[2 * l]; Wsh[l] = shapes[2 * l + 1];
            offs[l] = a; a += Hs[l] * Wsh[l];
        }
    }

    const int b  = pos[4 * n + 0];
    const int qi = pos[4 * n + 1];
    const int qj = pos[4 * n + 2];
    const int ql = pos[4 * n + 3];

    // (a) rotated query: thread t -> (head h, dim d)
    {
        const int h = t / HDIM, d = t % HDIM;
        const int dd = (d < HHALF) ? d : d - HHALF;
        float ang;
        if      (dd < NFREQ)     ang = (float)qi * powf(10.0f, -(float)dd / 8.0f);
        else if (dd < 2 * NFREQ) ang = (float)qj * powf(10.0f, -(float)(dd - 8) / 8.0f);
        else                     ang = (float)ql * powf(0.1f,  -(float)(dd - 16) / 8.0f);
        const float c = cosf(ang), s = sinf(ang);
        const float q1 = qproj[(size_t)n * DIM + h * HDIM + dd];
        const float q2 = qproj[(size_t)n * DIM + h * HDIM + HHALF + dd];
        s_qrot[t] = (d < HHALF) ? (q1 * c - q2 * s) : (q1 * s + q2 * c);
    }

    // key geometry
    if (t < NKEY) {
        const int kk = t;
        const int l  = kk / (NSW * NSW);
        const int r  = kk % (NSW * NSW);
        const int di = r / NSW - NSW / 2;
        const int dj = r % NSW - NSW / 2;
        const int ci = (int)floorf(((float)qi + 0.5f) * (float)Hs[l]  / (float)Hs[ql]);
        const int cj = (int)floorf(((float)qj + 0.5f) * (float)Wsh[l] / (float)Wsh[ql]);
        const int ii = ci + di, jj = cj + dj;
        const bool valid = (ii >= 0) && (ii < Hs[l]) && (jj >= 0) && (jj < Wsh[l]);
        const int iic = min(max(ii, 0), Hs[l] - 1);
        const int jjc = min(max(jj, 0), Wsh[l] - 1);
        s_flat[kk] = b * TOTAL + offs[l] + iic * Wsh[l] + jjc;
        s_mask[kk] = valid ? 0.0f : -1.0e9f;
        s_kif[kk]  = (float)iic;
        s_kjf[kk]  = (float)jjc;
    }
    __syncthreads();

    // (b) RoPE cos/sin table for all keys: NKEY*HHALF = 4704 entries
    for (int idx = t; idx < NKEY * HHALF; idx += 384) {
        const int kk = idx / HHALF;
        const int dd = idx % HHALF;
        float base, f;
        if      (dd < NFREQ)     { base = s_kif[kk]; f = powf(10.0f, -(float)dd / 8.0f); }
        else if (dd < 2 * NFREQ) { base = s_kjf[kk]; f = powf(10.0f, -(float)(dd - 8) / 8.0f); }
        else                     { base = (float)(kk / (NSW * NSW)); f = powf(0.1f, -(float)(dd - 16) / 8.0f); }
        const float ang = base * f;
        s_cos[idx] = cosf(ang);
        s_sin[idx] = sinf(ang);
    }
    __syncthreads();

    // (c) scores: 8 heads x 196 keys = 1568 dot products of length 48
    const float scale = rsqrtf((float)HDIM);
    for (int idx = t; idx < NHD * NKEY; idx += 384) {
        const int h  = idx / NKEY;
        const int kk = idx % NKEY;
        const float* kp = kvmaps + (size_t)s_flat[kk] * TWOD + h * HDIM;
        const float* cw = s_cos + kk * HHALF;
        const float* sw = s_sin + kk * HHALF;
        const float* qh = s_qrot + h * HDIM;
        float sc = 0.f;
#pragma unroll
        for (int dd = 0; dd < HHALF; ++dd) {
            const float k1 = kp[dd], k2 = kp[dd + HHALF];
            const float c = cw[dd], s = sw[dd];
            sc += qh[dd] * (k1 * c - k2 * s) + qh[dd + HHALF] * (k1 * s + k2 * c);
        }
        s_sc[idx] = sc * scale + s_mask[kk];
    }
    __syncthreads();

    // (d) softmax: wave w (w<8) handles head w, lanes stride keys
    if (t < NHD * 32) {
        const int h = t >> 5, lane = t & 31;
        float m = -3.4e38f;
        for (int kk = lane; kk < NKEY; kk += 32) m = fmaxf(m, s_sc[h * NKEY + kk]);
#pragma unroll
        for (int off = 16; off; off >>= 1) m = fmaxf(m, __shfl_xor(m, off, 32));
        float sum = 0.f;
        for (int kk = lane; kk < NKEY; kk += 32) {
            const float e = expf(s_sc[h * NKEY + kk] - m);
            s_sc[h * NKEY + kk] = e;
            sum += e;
        }
#pragma unroll
        for (int off = 16; off; off >>= 1) sum += __shfl_xor(sum, off, 32);
        const float inv = 1.f / sum;
        for (int kk = lane; kk < NKEY; kk += 32) s_sc[h * NKEY + kk] *= inv;
    }
    __syncthreads();

    // (e) output: thread t owns (h,d); 384 consecutive lanes read 4B-contiguous v
    {
        const int h = t / HDIM, d = t % HDIM;
        float acc = 0.f;
        for (int kk = 0; kk < NKEY; ++kk) {
            const float wgt = s_sc[h * NKEY + kk];
            const float v = kvmaps[(size_t)s_flat[kk] * TWOD + DIM + h * HDIM + d];
            acc = fmaf(wgt, v, acc);
        }
        attnout[(size_t)n * DIM + t] = acc;
    }
}

// ---------------------------------------------------------------------------
// Launch: LN -> q GEMM -> kv GEMM -> attention -> out GEMM (+residual)
// Workspace (floats): x_norm | q_proj | attn_out | kv_maps  ~= 71.6 MB
// ---------------------------------------------------------------------------
extern "C" void kernel_launch(void* const* d_in, const int* in_sizes, int n_in,
                              void* d_out, int out_size, void* d_ws, size_t ws_size,
                              hipStream_t stream)
{
    const float* query  = (const float*)d_in[0];
    const int*   pos    = (const int*)  d_in[1];
    // d_in[2] (batch offsets) unused: batch index comes from pos[:,0]
    const float* fmaps  = (const float*)d_in[3];
    const int*   shapes = (const int*)  d_in[4];
    const float* nw     = (const float*)d_in[5];
    const float* nb     = (const float*)d_in[6];
    const float* wq     = (const float*)d_in[7];
    const float* wkv    = (const float*)d_in[8];
    const float* wout   = (const float*)d_in[9];
    float* out = (float*)d_out;

    float* ws      = (float*)d_ws;
    float* x_norm  = ws;                       // N_Q*DIM
    float* q_proj  = x_norm + (size_t)N_Q * DIM;
    float* attn_o  = q_proj + (size_t)N_Q * DIM;
    float* kv_maps = attn_o + (size_t)N_Q * DIM;  // MROWS_KV*TWOD

    // 1. LayerNorm
    ln_kernel<<<N_Q / 8, 256, 0, stream>>>(query, nw, nb, x_norm);

    // 2. q = x_norm @ w_q^T       (1024 x 384 x 384)
    {
        dim3 grid(N_Q / 128, DIM / 64);
        wmma_gemm_nt_kernel<<<grid, 256, 0, stream>>>(x_norm, wq, nullptr, q_proj,
                                                      N_Q, DIM);
    }

    // 3. kv_maps = feature_maps @ w_kv^T   (21760 x 768 x 384) — dominant GEMM
    {
        dim3 grid(MROWS_KV / 128, TWOD / 64);
        wmma_gemm_nt_kernel<<<grid, 256, 0, stream>>>(fmaps, wkv, nullptr, kv_maps,
                                                      MROWS_KV, TWOD);
    }

    // 4. attention (gather + RoPE + softmax + weighted sum)
    attn_kernel<<<N_Q, 384, 0, stream>>>(q_proj, kv_maps, pos, shapes, attn_o);

    // 5. out = residual + attn_o @ w_out^T  (1024 x 384 x 384)
    {
        dim3 grid(N_Q / 128, DIM / 64);
        wmma_gemm_nt_kernel<<<grid, 256, 0, stream>>>(attn_o, wout, query, out,
                                                      N_Q, DIM);
    }
}